// FGCN_42331197669658
// MI455X (gfx1250) — compile-verified
//
#include <hip/hip_runtime.h>
#include <hip/hip_bf16.h>

#define B_      16
#define MAXLEN_ 1024
#define PE_     256
#define N_      1024
#define EMB_    256
#define GO_OUT_ 64
#define C_      128
#define HID_    64
#define EPS_    1e-5f

// Use CDNA5 async global->LDS copies (ASYNCcnt) for GEMM A-tile staging.
#define ASYNC_STAGE 1

typedef __attribute__((ext_vector_type(2))) float v2f;
typedef __attribute__((ext_vector_type(8))) float v8f;

// V_WMMA_F32_16X16X4_F32: A 16x4 f32 (2 VGPR), B 4x16 f32 (2 VGPR), C/D 16x16 f32 (8 VGPR)
// A: lane L, reg r -> A[M = L%16][K = r + 2*(L/16)]
// B: lane L, reg r -> B[K = r + 2*(L/16)][N = L%16]
// C: lane L, reg r -> C[M = r + 8*(L/16)][N = L%16]
__device__ __forceinline__ v8f wmma4(v2f a, v2f b, v8f c) {
  return __builtin_amdgcn_wmma_f32_16x16x4_f32(
      /*neg_a=*/false, a, /*neg_b=*/false, b,
      /*c_mod=*/(short)0, c, /*reuse_a=*/false, /*reuse_b=*/false);
}

// ---------------- Kernel 1: embedding mean + pe, cat = [feat | inp2+pe] (16 x 512)
__global__ void k_embed(const int* __restrict__ data, const float* __restrict__ emb,
                        float* __restrict__ cat) {
  int b = blockIdx.x;   // 16
  int e = threadIdx.x;  // 256
  const int row = b * (MAXLEN_ + PE_);
  float acc = 0.f;
  for (int t = 0; t < MAXLEN_; ++t) {
    int tok = data[row + t];
    acc += emb[(size_t)tok * EMB_ + e];
  }
  float pe = __sinf((float)e);
  cat[b * 512 + e]       = acc * (1.0f / MAXLEN_) + pe;
  cat[b * 512 + PE_ + e] = (float)data[row + MAXLEN_ + e] + pe;
}

// ---------------- Kernel 2: batchnorm over axis 0 (16 rows), 512 features
__global__ void k_bnfeat(const float* __restrict__ cat, const float* __restrict__ g1,
                         const float* __restrict__ b1, float* __restrict__ m1) {
  int j = blockIdx.x * blockDim.x + threadIdx.x;
  if (j >= 2 * PE_) return;
  float s = 0.f, sq = 0.f;
  for (int i = 0; i < B_; ++i) { float v = cat[i * 512 + j]; s += v; sq += v * v; }
  float mean = s * (1.0f / B_);
  float var  = sq * (1.0f / B_) - mean * mean;
  float sc = rsqrtf(var + EPS_) * g1[j];
  float sh = b1[j] - mean * sc;
  for (int i = 0; i < B_; ++i) m1[i * 512 + j] = cat[i * 512 + j] * sc + sh;
}

// ---------------- Kernel 3: go = relu(m1 @ w_go + b_go) scattered into x[:, :, 0:64]
// M=16, K=512, Ncols=65536. One wave = one 16x16 output tile; 4096 waves.
// w_go stream (134 MB) is the dominant HBM traffic; B loads are coalesced row
// segments, A (32 KB) is L2-resident.
__global__ void k_go(const float* __restrict__ m1, const float* __restrict__ wgo,
                     const float* __restrict__ bgo, float* __restrict__ x) {
  int wave = blockIdx.x * (blockDim.x >> 5) + (threadIdx.x >> 5);
  int lane = threadIdx.x & 31;
  int lh = lane >> 4, lm = lane & 15;
  int j0 = wave * 16;
  v8f c = {};
  for (int k0 = 0; k0 < 2 * PE_; k0 += 4) {
    int ka = k0 + 2 * lh;
    v2f a, b;
    a.x = m1[lm * 512 + ka];
    a.y = m1[lm * 512 + ka + 1];
    b.x = wgo[(size_t)ka * 65536 + j0 + lm];
    b.y = wgo[(size_t)(ka + 1) * 65536 + j0 + lm];
    c = wmma4(a, b, c);
  }
  int j = j0 + lm;
  int n = j >> 6, o = j & 63;
  float bias = bgo[j];
  union { v8f v; float f[8]; } u; u.v = c;
  for (int r = 0; r < 8; ++r) {
    int i = r + 8 * lh;  // batch row
    float v = u.f[r] + bias;
    v = v > 0.f ? v : 0.f;
    x[((size_t)(i * N_ + n)) * C_ + o] = v;
  }
}

// ---------------- Kernel 4: broadcast node_features into x[:, :, 64:128] (float4)
__global__ void k_nf(const float* __restrict__ nf, float* __restrict__ x) {
  int idx = blockIdx.x * blockDim.x + threadIdx.x;  // B*N*16 float4s = 262144
  int o4 = idx & 15;
  int n  = (idx >> 4) & (N_ - 1);
  int b  = idx >> 14;
  const float4 v = *(const float4*)&nf[n * 64 + o4 * 4];
  *(float4*)&x[((size_t)(b * N_ + n)) * C_ + GO_OUT_ + o4 * 4] = v;
}

// ---------------- Kernel 5: per-node batchnorm over (B, C), in place
__global__ void k_bn_node(float* __restrict__ x, const float* __restrict__ g,
                          const float* __restrict__ bb) {
  __shared__ float ss[256], sq2[256];
  int n = blockIdx.x;
  int t = threadIdx.x;
  float s = 0.f, sq = 0.f;
  for (int idx = t; idx < B_ * C_; idx += 256) {
    int b = idx >> 7, cc = idx & 127;
    float v = x[((size_t)(b * N_ + n)) * C_ + cc];
    s += v; sq += v * v;
  }
  ss[t] = s; sq2[t] = sq;
  __syncthreads();
  for (int o = 128; o > 0; o >>= 1) {
    if (t < o) { ss[t] += ss[t + o]; sq2[t] += sq2[t + o]; }
    __syncthreads();
  }
  float mean = ss[0] * (1.0f / (B_ * C_));
  float var  = sq2[0] * (1.0f / (B_ * C_)) - mean * mean;
  float sc = rsqrtf(var + EPS_) * g[n];
  float sh = bb[n] - mean * sc;
  for (int idx = t; idx < B_ * C_; idx += 256) {
    int b = idx >> 7, cc = idx & 127;
    size_t p = ((size_t)(b * N_ + n)) * C_ + cc;
    x[p] = x[p] * sc + sh;
  }
}

// ---------------- Tiled batched fp32 WMMA GEMM.
// Block = 128 threads = 4 waves. All 4 waves share one 16-row A strip; the
// strip is staged into LDS in 64-wide K chunks using CDNA5 async global->LDS
// b128 copies (ASYNCcnt + s_wait_asynccnt), then read back as WMMA A
// fragments via ds_load_b64 (pitch 68 floats: 16B-aligned rows, conflict-free).
// Wave w computes column tile j0 = (jg*4 + w)*16.
// C[b] = alpha * A[b] @ op(B[b]) [+ beta*S] [+ C (accum)]
__global__ void k_gemm_t(const float* __restrict__ A, const float* __restrict__ Bm,
                         float* __restrict__ Cm, const float* __restrict__ S,
                         int tilesM, int tilesN, int Kd,
                         int lda, int ldb, int ldc,
                         long strideA, long strideB, long strideC,
                         float alpha, float beta, int transB, int accum) {
  __shared__ __align__(16) float tileA[16][68];
  int jg4 = tilesN >> 2;
  int blk = blockIdx.x;
  int batch = blk / (tilesM * jg4);
  int rem = blk - batch * (tilesM * jg4);
  int i0 = (rem / jg4) * 16;
  int jg = rem % jg4;
  int wid = (threadIdx.x >> 5) & 3;
  int lane = threadIdx.x & 31;
  int lh = lane >> 4, lm = lane & 15;
  int j0 = (jg * 4 + wid) * 16;
  const float* Ab = A + (size_t)batch * strideA;
  const float* Bb = Bm + (size_t)batch * strideB;
  const float* ta = &tileA[0][0];
  v8f c = {};
  for (int kc = 0; kc < Kd; kc += 64) {
    // Stage A[i0:i0+16, kc:kc+64]: 256 x 16B transfers, 2 per thread, coalesced.
    for (int s2 = threadIdx.x; s2 < 256; s2 += 128) {
      int r = s2 >> 4;
      int c4 = (s2 & 15) << 2;
      const float* src = Ab + (size_t)(i0 + r) * lda + kc + c4;
#if ASYNC_STAGE
      unsigned int ldsoff = (unsigned int)(size_t)&tileA[r][c4];  // low 32 bits = LDS offset
      unsigned long long ga = (unsigned long long)(size_t)src;
      asm volatile("global_load_async_to_lds_b128 %0, %1, off"
                   :: "v"(ldsoff), "v"(ga) : "memory");
#else
      *(float4*)&tileA[r][c4] = *(const float4*)src;
#endif
    }
#if ASYNC_STAGE
    asm volatile("s_wait_asynccnt 0x0" ::: "memory");
#endif
    __syncthreads();
    for (int kl = 0; kl < 64; kl += 4) {
      int ka = kl + 2 * lh;
      int kg = kc + ka;
      v2f a, b;
      a.x = ta[lm * 68 + ka];
      a.y = ta[lm * 68 + ka + 1];
      if (transB) {
        b.x = Bb[(size_t)(j0 + lm) * ldb + kg];
        b.y = Bb[(size_t)(j0 + lm) * ldb + kg + 1];
      } else {
        b.x = Bb[(size_t)kg * ldb + j0 + lm];
        b.y = Bb[(size_t)(kg + 1) * ldb + j0 + lm];
      }
      c = wmma4(a, b, c);
    }
    __syncthreads();
  }
  union { v8f v; float f[8]; } u; u.v = c;
  size_t cb = (size_t)batch * strideC;
  for (int r = 0; r < 8; ++r) {
    int i = i0 + r + 8 * lh;
    size_t p = cb + (size_t)i * ldc + j0 + lm;
    float v = alpha * u.f[r];
    if (S)     v += beta * S[p];
    if (accum) v += Cm[p];
    Cm[p] = v;
  }
}

// ---------------- Softmax over axis=1 (n) per (b, m) column; adj = go_adj + softmax
__global__ void k_soft(float* __restrict__ att, const float* __restrict__ go_adj) {
  int b = blockIdx.x >> 2;
  int m = ((blockIdx.x & 3) << 8) + threadIdx.x;
  float* A = att + (size_t)b * N_ * N_;
  float mx = -3.4e38f;
  for (int n = 0; n < N_; ++n) { float v = A[(size_t)n * N_ + m]; mx = v > mx ? v : mx; }
  float s = 0.f;
  for (int n = 0; n < N_; ++n) s += __expf(A[(size_t)n * N_ + m] - mx);
  float inv = 1.0f / s;
  for (int n = 0; n < N_; ++n) {
    size_t p = (size_t)n * N_ + m;
    A[p] = go_adj[p] + __expf(A[p] - mx) * inv;
  }
}

// ---------------- Final: bn_node(out + x) -> relu -> @ w_out + b_out -> (B, N)
__global__ void k_final(const float* __restrict__ ob, const float* __restrict__ x,
                        const float* __restrict__ g4, const float* __restrict__ b4,
                        const float* __restrict__ wout, const float* __restrict__ bout,
                        float* __restrict__ y) {
  __shared__ float ss[256], sq2[256];
  int n = blockIdx.x;
  int t = threadIdx.x;
  float s = 0.f, sq = 0.f;
  for (int idx = t; idx < B_ * C_; idx += 256) {
    int b = idx >> 7, cc = idx & 127;
    size_t p = ((size_t)(b * N_ + n)) * C_ + cc;
    float v = ob[p] + x[p];
    s += v; sq += v * v;
  }
  ss[t] = s; sq2[t] = sq;
  __syncthreads();
  for (int o = 128; o > 0; o >>= 1) {
    if (t < o) { ss[t] += ss[t + o]; sq2[t] += sq2[t + o]; }
    __syncthreads();
  }
  float mean = ss[0] * (1.0f / (B_ * C_));
  float var  = sq2[0] * (1.0f / (B_ * C_)) - mean * mean;
  float sc = rsqrtf(var + EPS_) * g4[n];
  float sh = b4[n] - mean * sc;
  __syncthreads();
  int b = t >> 4;     // 0..15
  int part = t & 15;  // 0..15
  float acc = 0.f;
  for (int cc = part; cc < C_; cc += 16) {
    size_t p = ((size_t)(b * N_ + n)) * C_ + cc;
    float v = (ob[p] + x[p]) * sc + sh;
    v = v > 0.f ? v : 0.f;
    acc += v * wout[cc];
  }
  ss[t] = acc;
  __syncthreads();
  for (int o = 8; o > 0; o >>= 1) {
    if (part < o) ss[t] += ss[t + o];
    __syncthreads();
  }
  if (part == 0) y[b * N_ + n] = ss[t] + bout[0];
}

extern "C" void kernel_launch(void* const* d_in, const int* in_sizes, int n_in,
                              void* d_out, int out_size, void* d_ws, size_t ws_size,
                              hipStream_t stream) {
  const int*   data  = (const int*)d_in[0];
  const float* nodef = (const float*)d_in[1];
  const float* goadj = (const float*)d_in[2];
  // d_in[3] batch_adj unused
  const float* emb   = (const float*)d_in[4];
  const float* w_go  = (const float*)d_in[5];
  const float* b_go  = (const float*)d_in[6];
  const float* wq    = (const float*)d_in[7];
  const float* wk    = (const float*)d_in[8];
  const float* theta = (const float*)d_in[9];
  const float* g1    = (const float*)d_in[10];
  const float* b1    = (const float*)d_in[11];
  const float* g3    = (const float*)d_in[12];
  const float* b3    = (const float*)d_in[13];
  const float* g4    = (const float*)d_in[14];
  const float* b4    = (const float*)d_in[15];
  const float* w_out = (const float*)d_in[16];
  const float* b_out = (const float*)d_in[17];
  float* out = (float*)d_out;

  float* ws = (float*)d_ws;
  size_t o = 0;
  float* cat = ws + o; o += 16 * 512;
  float* m1  = ws + o; o += 16 * 512;
  float* x   = ws + o; o += (size_t)16 * 1024 * 128;   // (B,N,C)
  float* q   = ws + o; o += (size_t)16 * 1024 * 64;
  float* kk  = ws + o; o += (size_t)16 * 1024 * 64;
  float* att = ws + o; o += (size_t)16 * 1024 * 1024;  // reused as adj
  float* tc  = ws + o; o += (size_t)16 * 1024 * 128;
  float* tn  = ws + o; o += (size_t)16 * 1024 * 128;
  float* ob  = ws + o; o += (size_t)16 * 1024 * 128;

  k_embed  <<<16,   256, 0, stream>>>(data, emb, cat);
  k_bnfeat <<<2,    256, 0, stream>>>(cat, g1, b1, m1);
  k_go     <<<1024, 128, 0, stream>>>(m1, w_go, b_go, x);          // 4096 waves
  k_nf     <<<1024, 256, 0, stream>>>(nodef, x);
  k_bn_node<<<1024, 256, 0, stream>>>(x, g3, b3);
  // q = x2d(16384x128) @ wq(128x64): tilesM=1024, tilesN=4 -> 1024 blocks
  k_gemm_t<<<1024, 128, 0, stream>>>(x, wq, q, nullptr, 1024, 4, 128, 128, 64, 64,
                                     0, 0, 0, 1.f, 0.f, 0, 0);
  k_gemm_t<<<1024, 128, 0, stream>>>(x, wk, kk, nullptr, 1024, 4, 128, 128, 64, 64,
                                     0, 0, 0, 1.f, 0.f, 0, 0);
  // att[b] = q[b](1024x64) @ k[b]^T / sqrt(64): 16*64*16 = 16384 blocks
  k_gemm_t<<<16384, 128, 0, stream>>>(q, kk, att, nullptr, 64, 64, 64, 64, 64, 1024,
                                      (long)1024 * 64, (long)1024 * 64, (long)1024 * 1024,
                                      0.125f, 0.f, 1, 0);
  k_soft<<<64, 256, 0, stream>>>(att, goadj);                      // att -> adj in place
  // t_cur = adj @ x: 16*64*2 = 2048 blocks
  k_gemm_t<<<2048, 128, 0, stream>>>(att, x, tc, nullptr, 64, 8, 1024, 1024, 128, 128,
                                     (long)1024 * 1024, (long)1024 * 128, (long)1024 * 128,
                                     1.f, 0.f, 0, 0);
  // t_next = 2 * adj @ t_cur - x
  k_gemm_t<<<2048, 128, 0, stream>>>(att, tc, tn, x, 64, 8, 1024, 1024, 128, 128,
                                     (long)1024 * 1024, (long)1024 * 128, (long)1024 * 128,
                                     2.f, -1.f, 0, 0);
  // out = x@theta0 + t_cur@theta1 + t_next@theta2: 1024*2 = 2048 blocks each
  k_gemm_t<<<2048, 128, 0, stream>>>(x,  theta,                ob, nullptr, 1024, 8, 128,
                                     128, 128, 128, 0, 0, 0, 1.f, 0.f, 0, 0);
  k_gemm_t<<<2048, 128, 0, stream>>>(tc, theta + 128 * 128,    ob, nullptr, 1024, 8, 128,
                                     128, 128, 128, 0, 0, 0, 1.f, 0.f, 0, 1);
  k_gemm_t<<<2048, 128, 0, stream>>>(tn, theta + 2 * 128 * 128, ob, nullptr, 1024, 8, 128,
                                     128, 128, 128, 0, 0, 0, 1.f, 0.f, 0, 1);
  k_final<<<1024, 256, 0, stream>>>(ob, x, g4, b4, w_out, b_out, out);

  (void)n_in; (void)in_sizes; (void)out_size; (void)ws_size;
}